// EquivariantMessagePasser_18468359373229
// MI455X (gfx1250) — compile-verified
//
#include <hip/hip_runtime.h>

typedef __attribute__((ext_vector_type(2))) float v2f;
typedef __attribute__((ext_vector_type(8))) float v8f;

#define N_ATOMS   10000
#define N_EDGES   160000
#define N_BASIS   16
#define CUTOFF    5.0f
#define MP_SCALING 0.1f

// Native CDNA5 f32 scatter-add: GLOBAL_ATOMIC_ADD_F32 (no return -> STOREcnt),
// device scope so accumulators are coherent across WGPs (they live in L2).
// Inline asm guarantees the hardware atomic instead of a CAS expansion.
__device__ __forceinline__ void atomicAddF32(float* addr, float v) {
    asm volatile("global_atomic_add_f32 %0, %1, off scope:SCOPE_DEV"
                 :: "v"(addr), "v"(v)
                 : "memory");
}

__global__ void zero_kernel(float* __restrict__ p, int n) {
    int i = blockIdx.x * blockDim.x + threadIdx.x;
    if (i < n) p[i] = 0.0f;
}

// -------------------------------------------------------------------------
// Edge kernel: one thread per (edge, channel k).
// Computes radial basis (Chebyshev recurrence), rad = basis @ Wr[:,k],
// CG contraction ce[p] = rad * sum_{m,n} sh[m]*src[n]*U[m,n,p],
// then atomic scatter-add into per-atom accumulators (fits in L2).
// -------------------------------------------------------------------------
template<int L, int M_, int K, int EPB>
__global__ __launch_bounds__(K * EPB)
void edge_kernel(const float* __restrict__ r,
                 const float* __restrict__ sh,
                 const int*   __restrict__ centers,
                 const int*   __restrict__ neighbors,
                 const float* __restrict__ fe,
                 const float* __restrict__ fo,
                 const float* __restrict__ Wr,
                 const float* __restrict__ U,
                 float* __restrict__ pe_ws,
                 float* __restrict__ po_ws) {
    __shared__ float sU[M_ * M_ * M_];
    __shared__ float sWr[N_BASIS * K];
    for (int i = threadIdx.x; i < M_ * M_ * M_; i += blockDim.x) sU[i] = U[i];
    for (int i = threadIdx.x; i < N_BASIS * K; i += blockDim.x)  sWr[i] = Wr[i];
    __syncthreads();

    const int k  = threadIdx.x % K;
    const int le = threadIdx.x / K;
    const int e  = blockIdx.x * EPB + le;
    if (e >= N_EDGES) return;

    // radial basis: basis_b = cos(rr*b) * fcut, via Chebyshev recurrence
    const float rv   = r[e];
    const float rr   = rv * CUTOFF;
    const float fcut = 0.5f * (cospif(rv) + 1.0f); // cos(pi*rr/CUTOFF) == cos(pi*r)
    const float c1   = __cosf(rr);
    const float two_c1 = 2.0f * c1;

    float rad = sWr[0 * K + k];            // b = 0 (cheb = 1)
    float cb_m1 = 1.0f, cb = c1;
    rad = fmaf(sWr[1 * K + k], cb, rad);   // b = 1
    #pragma unroll
    for (int b = 2; b < N_BASIS; ++b) {
        float cn = fmaf(two_c1, cb, -cb_m1);
        cb_m1 = cb; cb = cn;
        rad = fmaf(sWr[b * K + k], cb, rad);
    }
    rad *= fcut;

    float shl[M_];
    #pragma unroll
    for (int m = 0; m < M_; ++m) shl[m] = sh[e * M_ + m];

    const int nbr = neighbors[e];
    const int ctr = centers[e];

    float se[M_], so[M_];
    #pragma unroll
    for (int n = 0; n < M_; ++n) {
        const float g_e = fe[(nbr * M_ + n) * K + k];  // coalesced in k
        const float g_o = fo[(nbr * M_ + n) * K + k];
        if (L % 2 == 0) { se[n] = g_e; so[n] = g_o; }   // parity routing
        else            { se[n] = g_o; so[n] = g_e; }
    }

    float ace[M_] = {}, aco[M_] = {};
    #pragma unroll
    for (int m = 0; m < M_; ++m) {
        #pragma unroll
        for (int n = 0; n < M_; ++n) {
            const float te = shl[m] * se[n];
            const float to = shl[m] * so[n];
            #pragma unroll
            for (int p = 0; p < M_; ++p) {
                const float u = sU[(m * M_ + n) * M_ + p];
                ace[p] = fmaf(te, u, ace[p]);
                aco[p] = fmaf(to, u, aco[p]);
            }
        }
    }

    const float s = rad * MP_SCALING;
    #pragma unroll
    for (int p = 0; p < M_; ++p) {
        atomicAddF32(&pe_ws[(ctr * M_ + p) * K + k], ace[p] * s);
        atomicAddF32(&po_ws[(ctr * M_ + p) * K + k], aco[p] * s);
    }
}

// -------------------------------------------------------------------------
// WMMA GEMM: out[R,K] = f[R,K] + (pacc[R,K]) @ W[K,K]
// One wave32 computes a 16x16 f32 tile via V_WMMA_F32_16X16X4_F32,
// stepping the inner dimension in chunks of 4.
// A layout (16x4 f32): lanes 0-15 hold {K=0,K=1}, lanes 16-31 hold {K=2,K=3}.
// C/D layout: VGPR i -> rows i (lanes 0-15) / i+8 (lanes 16-31), col = lane&15.
// -------------------------------------------------------------------------
template<int K>
__global__ __launch_bounds__(256)
void gemm_kernel(const float* __restrict__ pacc,
                 const float* __restrict__ f,
                 const float* __restrict__ W,
                 float* __restrict__ out,
                 int R) {
    const int lane = threadIdx.x & 31;
    const int wave = (blockIdx.x * blockDim.x + threadIdx.x) >> 5;
    const int tiles_n = K / 16;
    const int tiles_r = R / 16;
    if (wave >= tiles_r * tiles_n) return;   // uniform per wave: EXEC all-1s for WMMA

    const int tr = wave / tiles_n;
    const int tn = wave % tiles_n;
    const int half = lane >> 4;              // 0: lanes 0-15, 1: lanes 16-31
    const int l16  = lane & 15;

    const int arow = tr * 16 + l16;          // A-matrix row owned by this lane
    const int bcol = tn * 16 + l16;          // B-matrix column owned by this lane

    v8f acc = {};
    #pragma unroll
    for (int kk = 0; kk < K; kk += 4) {
        const int ka = kk + half * 2;
        v2f a, b;
        a.x = pacc[arow * K + ka];
        a.y = pacc[arow * K + ka + 1];
        b.x = W[(ka    ) * K + bcol];
        b.y = W[(ka + 1) * K + bcol];
        acc = __builtin_amdgcn_wmma_f32_16x16x4_f32(
            /*neg_a=*/false, a, /*neg_b=*/false, b,
            /*c_mod=*/(short)0, acc, /*reuse_a=*/false, /*reuse_b=*/false);
    }

    #pragma unroll
    for (int i = 0; i < 8; ++i) {
        const int row = tr * 16 + (half ? (i + 8) : i);
        const int col = tn * 16 + l16;
        out[row * K + col] = f[row * K + col] + acc[i];
    }
}

extern "C" void kernel_launch(void* const* d_in, const int* in_sizes, int n_in,
                              void* d_out, int out_size, void* d_ws, size_t ws_size,
                              hipStream_t stream) {
    (void)in_sizes; (void)n_in; (void)out_size; (void)ws_size;

    const float* r         = (const float*)d_in[0];
    const float* sh0       = (const float*)d_in[1];
    const float* sh1       = (const float*)d_in[2];
    const float* sh2       = (const float*)d_in[3];
    const int*   centers   = (const int*)d_in[4];
    const int*   neighbors = (const int*)d_in[5];
    const float* fe0 = (const float*)d_in[6];
    const float* fo0 = (const float*)d_in[7];
    const float* fe1 = (const float*)d_in[8];
    const float* fo1 = (const float*)d_in[9];
    const float* fe2 = (const float*)d_in[10];
    const float* fo2 = (const float*)d_in[11];
    const float* Wr0 = (const float*)d_in[12];
    const float* U0  = (const float*)d_in[13];
    const float* We0 = (const float*)d_in[14];
    const float* Wo0 = (const float*)d_in[15];
    const float* Wr1 = (const float*)d_in[16];
    const float* U1  = (const float*)d_in[17];
    const float* We1 = (const float*)d_in[18];
    const float* Wo1 = (const float*)d_in[19];
    const float* Wr2 = (const float*)d_in[20];
    const float* U2  = (const float*)d_in[21];
    const float* We2 = (const float*)d_in[22];
    const float* Wo2 = (const float*)d_in[23];

    // workspace: per-atom message accumulators (29.4 MB, fits in L2)
    float* ws  = (float*)d_ws;
    float* pe0 = ws + 0;        // 10000*1*64 = 640000
    float* po0 = ws + 640000;
    float* pe1 = ws + 1280000;  // 10000*3*48 = 1440000
    float* po1 = ws + 2720000;
    float* pe2 = ws + 4160000;  // 10000*5*32 = 1600000
    float* po2 = ws + 5760000;
    const int WS_FLOATS = 7360000;

    zero_kernel<<<(WS_FLOATS + 255) / 256, 256, 0, stream>>>(ws, WS_FLOATS);

    edge_kernel<0, 1, 64, 4><<<N_EDGES / 4, 256, 0, stream>>>(
        r, sh0, centers, neighbors, fe0, fo0, Wr0, U0, pe0, po0);
    edge_kernel<1, 3, 48, 4><<<N_EDGES / 4, 192, 0, stream>>>(
        r, sh1, centers, neighbors, fe1, fo1, Wr1, U1, pe1, po1);
    edge_kernel<2, 5, 32, 8><<<N_EDGES / 8, 256, 0, stream>>>(
        r, sh2, centers, neighbors, fe2, fo2, Wr2, U2, pe2, po2);

    // outputs concatenated flat: oe0, oo0, oe1, oo1, oe2, oo2
    float* out = (float*)d_out;
    float* oe0 = out + 0;
    float* oo0 = out + 640000;
    float* oe1 = out + 1280000;
    float* oo1 = out + 2720000;
    float* oe2 = out + 4160000;
    float* oo2 = out + 5760000;

    auto blocks_for = [](int R, int K) {
        int waves = (R / 16) * (K / 16);
        return (waves * 32 + 255) / 256;
    };

    gemm_kernel<64><<<blocks_for(10000, 64), 256, 0, stream>>>(pe0, fe0, We0, oe0, 10000);
    gemm_kernel<64><<<blocks_for(10000, 64), 256, 0, stream>>>(po0, fo0, Wo0, oo0, 10000);
    gemm_kernel<48><<<blocks_for(30000, 48), 256, 0, stream>>>(pe1, fe1, We1, oe1, 30000);
    gemm_kernel<48><<<blocks_for(30000, 48), 256, 0, stream>>>(po1, fo1, Wo1, oo1, 30000);
    gemm_kernel<32><<<blocks_for(50000, 32), 256, 0, stream>>>(pe2, fe2, We2, oe2, 50000);
    gemm_kernel<32><<<blocks_for(50000, 32), 256, 0, stream>>>(po2, fo2, Wo2, oo2, 50000);
}